// EncoderNetWithLSTM_15547781612107
// MI455X (gfx1250) — compile-verified
//
#include <hip/hip_runtime.h>

// ---------------------------------------------------------------------------
// 2-layer LSTM single step, PB = 64*256 = 16384 fused batch, H = 512.
// f32 WMMA (V_WMMA_F32_16X16X4_F32) GEMMs; weight tiles double-buffered into
// LDS via GLOBAL_LOAD_ASYNC_TO_LDS_B128 (ASYNCcnt) when available; fused LSTM
// gate epilogue. Problem is memory-bound (~310 MB compulsory @ 23.3 TB/s), so
// the blocking below minimizes L2 re-read traffic: each wave computes a
// 16M x 32h tile (8 gate-tiles, 64 acc VGPRs) so each A fragment feeds 8
// WMMAs, and A rows are re-read by only H/32 = 16 blocks.
// ---------------------------------------------------------------------------

namespace {
constexpr int P = 64, B = 256, H = 512, L = 2;
constexpr int PB = P * B;                  // 16384 rows
constexpr int K = H;                       // GEMM depth per weight matrix
constexpr int KC = 64;                     // K chunk staged in LDS
constexpr int LDSS = 68;                   // padded LDS stride (floats) per col
constexpr int NCOL = 128;                  // weight cols per block: 4 gates x 32 h
constexpr int OUT_ELEMS = B * P * H;       // 8388608, size of `out`
constexpr int HN_OFF = OUT_ELEMS;          // start of hn [P,L,B,H]
constexpr int CN_OFF = HN_OFF + P * L * B * H;  // start of cn
}  // namespace

typedef __attribute__((ext_vector_type(2))) float v2f;
typedef __attribute__((ext_vector_type(8))) float v8f;
typedef int v4i __attribute__((vector_size(16)));

#if __has_builtin(__builtin_amdgcn_global_load_async_to_lds_b128) && \
    __has_builtin(__builtin_amdgcn_s_wait_asynccnt)
#define USE_ASYNC_LDS 1
#else
#define USE_ASYNC_LDS 0
#endif

__device__ __forceinline__ void copy16(const float* g, float* l) {
#if USE_ASYNC_LDS
  typedef __attribute__((address_space(1))) v4i* gptr_t;  // global (AS1) v4i*
  typedef __attribute__((address_space(3))) v4i* lptr_t;  // LDS (AS3) v4i*
  // Generic->AS pointers via integers: AS1 keeps the full 64-bit address,
  // AS3 keeps the low 32 bits (= LDS byte offset per the aperture mapping).
  __builtin_amdgcn_global_load_async_to_lds_b128(
      (gptr_t)(unsigned long long)(__SIZE_TYPE__)g,
      (lptr_t)(unsigned int)(__SIZE_TYPE__)l, 0, 0);
#else
  *(float4*)l = *(const float4*)g;
#endif
}

__device__ __forceinline__ void async_wait_all() {
#if USE_ASYNC_LDS
  __builtin_amdgcn_s_wait_asynccnt(0);
#endif
}

__device__ __forceinline__ float sigm(float x) { return 1.0f / (1.0f + __expf(-x)); }
__device__ __forceinline__ float tanh_fast(float x) {
  return 2.0f / (1.0f + __expf(-2.0f * x)) - 1.0f;
}

// Issue the async fill of one 128(cols) x 64(K) weight tile into `buf`.
// Column c (0..127) covers global row n = (c>>5)*H + hbase + (c&31) of W.
__device__ __forceinline__ void issue_fill(const float* __restrict__ W, int hbase,
                                           int kc, float* buf) {
  const int t = threadIdx.x;
  const int c = t >> 1;                // 0..127
  const int kq = (t & 1) * 32;         // this thread moves 32 K values
  const int n = (c >> 5) * H + hbase + (c & 31);
  const float* g = W + n * K + kc + kq;
  float* l = buf + c * LDSS + kq;
#pragma unroll
  for (int i = 0; i < 8; ++i) copy16(g + 4 * i, l + 4 * i);
}

// acc[j] += A_tile(16 x 512) x W_tile(512 x 16) for 8 column tiles j.
// Tile j covers W rows n = (j>>1)*H + hbase + (j&1)*16 + lanelo.
// arow: per-lane pointer to this lane's A row (row-major, stride K).
__device__ __forceinline__ void gemm_k512(const float* __restrict__ arow,
                                          const float* __restrict__ W,
                                          int hbase, int lanelo, int koff,
                                          float (*lds)[NCOL * LDSS], v8f acc[8]) {
  issue_fill(W, hbase, 0, lds[0]);
  int ib = 0;
  for (int kc = 0; kc < K; kc += KC, ib ^= 1) {
    async_wait_all();                  // my fill of lds[ib] complete
    __syncthreads();                   // everyone's fill complete / prev reads done
    if (kc + KC < K) issue_fill(W, hbase, kc + KC, lds[ib ^ 1]);  // overlap
    const float* abase = arow + kc + koff;
    const float* buf = lds[ib];
#pragma unroll 2
    for (int kk = 0; kk < KC; kk += 4) {
      const float2 av = *(const float2*)(abase + kk);   // lane's 2 K-values
      v2f af;
      af.x = av.x;
      af.y = av.y;
#pragma unroll
      for (int j = 0; j < 8; ++j) {
        const float2 bv =
            *(const float2*)(buf + (j * 16 + lanelo) * LDSS + kk + koff);
        v2f bf;
        bf.x = bv.x;
        bf.y = bv.y;
        acc[j] = __builtin_amdgcn_wmma_f32_16x16x4_f32(
            false, af, false, bf, (short)0, acc[j], false, false);
      }
    }
  }
}

// ------------------------------- layer 0 ----------------------------------
__global__ __launch_bounds__(256) void lstm_layer0(
    const float* __restrict__ xin,   // input_traces [B,P,2]
    const float* __restrict__ h0,    // [P,L,B,H]
    const float* __restrict__ c0,    // [P,L,B,H]
    const float* __restrict__ wih0,  // [4H,2]
    const float* __restrict__ whh0,  // [4H,H]
    const float* __restrict__ bih0, const float* __restrict__ bhh0,
    float* __restrict__ out) {       // full d_out (out|hn|cn)
  __shared__ float lds[2][NCOL * LDSS];
  const int lane = threadIdx.x & 31;
  const int wv = threadIdx.x >> 5;
  const int lanelo = lane & 15;
  const int lanehi = lane >> 4;
  const int koff = lanehi * 2;
  const int mbase = blockIdx.x * 128 + wv * 16;
  const int hbase = blockIdx.y * 32;

  const int mrow = mbase + lanelo;
  const int pr = mrow >> 8, br = mrow & 255;
  const float* arow = h0 + ((pr * L + 0) * B + br) * H;

  v8f acc[8] = {};
  gemm_k512(arow, whh0, hbase, lanelo, koff, lds, acc);

  // Fused epilogue: rank-2 x-update + biases + LSTM nonlinearity.
#pragma unroll
  for (int half = 0; half < 2; ++half) {
    const int hc = hbase + half * 16 + lanelo;
    float wxa[4], wxb[4], bsum[4];
#pragma unroll
    for (int g = 0; g < 4; ++g) {
      wxa[g] = wih0[(g * H + hc) * 2 + 0];
      wxb[g] = wih0[(g * H + hc) * 2 + 1];
      bsum[g] = bih0[g * H + hc] + bhh0[g * H + hc];
    }
#pragma unroll
    for (int r = 0; r < 8; ++r) {
      const int m = mbase + lanehi * 8 + r;     // acc row -> global row
      const int p = m >> 8, b = m & 255;
      const float x0 = xin[(b * P + p) * 2 + 0];
      const float x1 = xin[(b * P + p) * 2 + 1];
      const float gi = acc[0 + half][r] + x0 * wxa[0] + x1 * wxb[0] + bsum[0];
      const float gf = acc[2 + half][r] + x0 * wxa[1] + x1 * wxb[1] + bsum[1];
      const float gg = acc[4 + half][r] + x0 * wxa[2] + x1 * wxb[2] + bsum[2];
      const float go = acc[6 + half][r] + x0 * wxa[3] + x1 * wxb[3] + bsum[3];
      const int rowc = ((p * L + 0) * B + b) * H + hc;
      const float cp = c0[rowc];
      const float ig = sigm(gi), fg = sigm(gf), gt = tanh_fast(gg), og = sigm(go);
      const float cn = fg * cp + ig * gt;
      const float hn = og * tanh_fast(cn);
      out[HN_OFF + rowc] = hn;   // hn[p,0,b,hc]  (also layer-1 x input)
      out[CN_OFF + rowc] = cn;   // cn[p,0,b,hc]
    }
  }
}

// ------------------------------- layer 1 ----------------------------------
__global__ __launch_bounds__(256) void lstm_layer1(
    const float* __restrict__ h0, const float* __restrict__ c0,
    const float* __restrict__ wih1,  // [4H,H]
    const float* __restrict__ whh1,  // [4H,H]
    const float* __restrict__ bih1, const float* __restrict__ bhh1,
    float* __restrict__ out) {
  __shared__ float lds[2][NCOL * LDSS];
  const int lane = threadIdx.x & 31;
  const int wv = threadIdx.x >> 5;
  const int lanelo = lane & 15;
  const int lanehi = lane >> 4;
  const int koff = lanehi * 2;
  const int mbase = blockIdx.x * 128 + wv * 16;
  const int hbase = blockIdx.y * 32;

  const int mrow = mbase + lanelo;
  const int pr = mrow >> 8, br = mrow & 255;
  // x1 = layer-0 h_new, read back from the hn[:,0] slice written by layer 0.
  const float* arow_x = out + HN_OFF + ((pr * L + 0) * B + br) * H;
  const float* arow_h = h0 + ((pr * L + 1) * B + br) * H;

  v8f acc[8] = {};
  gemm_k512(arow_x, wih1, hbase, lanelo, koff, lds, acc);
  gemm_k512(arow_h, whh1, hbase, lanelo, koff, lds, acc);

#pragma unroll
  for (int half = 0; half < 2; ++half) {
    const int hc = hbase + half * 16 + lanelo;
    float bsum[4];
#pragma unroll
    for (int g = 0; g < 4; ++g) bsum[g] = bih1[g * H + hc] + bhh1[g * H + hc];
#pragma unroll
    for (int r = 0; r < 8; ++r) {
      const int m = mbase + lanehi * 8 + r;
      const int p = m >> 8, b = m & 255;
      const float gi = acc[0 + half][r] + bsum[0];
      const float gf = acc[2 + half][r] + bsum[1];
      const float gg = acc[4 + half][r] + bsum[2];
      const float go = acc[6 + half][r] + bsum[3];
      const int rowc = ((p * L + 1) * B + b) * H + hc;
      const float cp = c0[rowc];
      const float ig = sigm(gi), fg = sigm(gf), gt = tanh_fast(gg), og = sigm(go);
      const float cn = fg * cp + ig * gt;
      const float hn = og * tanh_fast(cn);
      out[HN_OFF + rowc] = hn;             // hn[p,1,b,hc]
      out[CN_OFF + rowc] = cn;             // cn[p,1,b,hc]
      out[(b * P + p) * H + hc] = hn;      // out[b,p,hc]
    }
  }
}

// ---------------------------------------------------------------------------
extern "C" void kernel_launch(void* const* d_in, const int* in_sizes, int n_in,
                              void* d_out, int out_size, void* d_ws,
                              size_t ws_size, hipStream_t stream) {
  const float* xin  = (const float*)d_in[0];   // input_traces [B,P,2]
  const float* h0   = (const float*)d_in[1];   // [P,L,B,H]
  const float* c0   = (const float*)d_in[2];   // [P,L,B,H]
  const float* wih0 = (const float*)d_in[3];
  const float* whh0 = (const float*)d_in[4];
  const float* bih0 = (const float*)d_in[5];
  const float* bhh0 = (const float*)d_in[6];
  const float* wih1 = (const float*)d_in[7];
  const float* whh1 = (const float*)d_in[8];
  const float* bih1 = (const float*)d_in[9];
  const float* bhh1 = (const float*)d_in[10];
  float* out = (float*)d_out;

  dim3 grid(PB / 128, H / 32);   // (128, 16): 128-row x 32-hcol tiles
  dim3 blk(256);                 // 8 wave32s, stacked along M
  lstm_layer0<<<grid, blk, 0, stream>>>(xin, h0, c0, wih0, whh0, bih0, bhh0, out);
  lstm_layer1<<<grid, blk, 0, stream>>>(h0, c0, wih1, whh1, bih1, bhh1, out);
}